// GraphSAGE_34548716929127
// MI455X (gfx1250) — compile-verified
//
#include <hip/hip_runtime.h>

// CDNA5 wave32 WMMA vector types
typedef __attribute__((ext_vector_type(2))) float v2f;
typedef __attribute__((ext_vector_type(8))) float v8f;

// ---------------------------------------------------------------------------
// Zero-fill (d_ws is poisoned with 0xAA; accumulators must start at 0)
// ---------------------------------------------------------------------------
__global__ void zero_f4_kernel(float4* __restrict__ p, long long n4) {
  long long i = (long long)blockIdx.x * blockDim.x + threadIdx.x;
  if (i < n4) {
    float4 z; z.x = 0.f; z.y = 0.f; z.z = 0.f; z.w = 0.f;
    p[i] = z;
  }
}

// ---------------------------------------------------------------------------
// In-degree via float atomics (deg fits in L2; 1.6M atomics, trivial)
// ---------------------------------------------------------------------------
__global__ void degree_kernel(const long long* __restrict__ dst,
                              float* __restrict__ deg, long long nE) {
  long long i = (long long)blockIdx.x * blockDim.x + threadIdx.x;
  if (i < nE) atomicAdd(deg + dst[i], 1.0f);
}

// ---------------------------------------------------------------------------
// Edge scatter-add: msg[dst] += x[src].  One thread per (edge, float4-chunk).
// x / msg both fit in the 192MB L2 -> this is L2-resident atomic traffic.
// C4 = channels/4 (compile-time so index math is shifts).
// ---------------------------------------------------------------------------
template <int C4>
__global__ void scatter_kernel(const float* __restrict__ xin,
                               const long long* __restrict__ src,
                               const long long* __restrict__ dst,
                               float* __restrict__ msg, long long nE) {
  long long t = (long long)blockIdx.x * blockDim.x + threadIdx.x;
  if (t >= nE * C4) return;
  long long e = t / C4;
  int c = (int)(t - e * C4);
  long long s = src[e];
  long long d = dst[e];
  const float4 v = ((const float4*)(xin + s * (long long)(C4 * 4)))[c];
  float* m = msg + d * (long long)(C4 * 4) + (long long)c * 4;
  atomicAdd(m + 0, v.x);
  atomicAdd(m + 1, v.y);
  atomicAdd(m + 2, v.z);
  atomicAdd(m + 3, v.w);
}

// ---------------------------------------------------------------------------
// Fused SAGE layer: out[nodes, COUT] =
//     (msg/deg) @ Wl^T  +  xin @ Wr^T  + bias   (optional ReLU)
//
// One wave32 owns a 16x16 (nodes x out-ch) tile using V_WMMA_F32_16X16X4_F32
// (full f32 precision — GEMM is <1% of runtime, aggregation dominates).
//
// ISA VGPR layouts (CDNA5 ISA, 7.12.2):
//   A 16x4 f32 : lanes 0-15 -> M=lane (K pair 0,1); lanes 16-31 -> M=lane-16
//                (K pair 2,3).  a.x=K+0, a.y=K+1 of the lane's pair.
//   B 4x16 f32 : lanes 0-15 -> N=lane (K pair 0,1); lanes 16-31 -> K pair 2,3.
//   C/D 16x16  : 8 VGPRs; vgpr i = row M=i (lanes 0-15) / M=8+i (lanes 16-31),
//                N = lane&15.
// ---------------------------------------------------------------------------
template <int K, int COUT, bool RELU>
__global__ __launch_bounds__(256) void sage_gemm_kernel(
    const float* __restrict__ msg,   // [n, K] summed neighbor messages
    const float* __restrict__ deg,   // [n]
    const float* __restrict__ xin,   // [n, K] self features
    const float* __restrict__ Wl,    // [COUT, K] row-major
    const float* __restrict__ Wr,    // [COUT, K] row-major
    const float* __restrict__ bias,  // [COUT]
    float* __restrict__ out,         // [n, COUT]
    long long n) {
  const int wave = threadIdx.x >> 5;                 // 8 waves / workgroup
  const int lane = threadIdx.x & 31;
  const long long tilesN = COUT >> 4;
  const long long numTiles = (n >> 4) * tilesN;
  const long long tile = (long long)blockIdx.x * 8 + wave;
  if (tile >= numTiles) return;                      // wave-uniform: EXEC stays all-1s

  const int tn = (int)(tile % tilesN);               // out-channel tile
  const long long tm = tile / tilesN;                // node tile

  const int l15 = lane & 15;
  const int hi  = lane >> 4;                         // 0 -> K pair (0,1); 1 -> (2,3)
  const long long rowA = tm * 16 + l15;              // node row this lane feeds
  const int colB = tn * 16 + l15;                    // out channel this lane feeds

  const float rdeg = 1.0f / fmaxf(deg[rowA], 1.0f);

  const float* __restrict__ msgRow = msg + rowA * K;
  const float* __restrict__ xRow   = xin + rowA * K;
  const float* __restrict__ wlRow  = Wl + (long long)colB * K;
  const float* __restrict__ wrRow  = Wr + (long long)colB * K;

  v8f acc = {};

  // D += mean @ Wl^T
#pragma unroll 8
  for (int k0 = 0; k0 < K; k0 += 4) {
    const int ka = k0 + hi * 2;
    v2f a, b;
    a.x = msgRow[ka]     * rdeg;
    a.y = msgRow[ka + 1] * rdeg;
    b.x = wlRow[ka];
    b.y = wlRow[ka + 1];
    acc = __builtin_amdgcn_wmma_f32_16x16x4_f32(false, a, false, b,
                                                (short)0, acc, false, false);
  }
  // D += x @ Wr^T
#pragma unroll 8
  for (int k0 = 0; k0 < K; k0 += 4) {
    const int ka = k0 + hi * 2;
    v2f a, b;
    a.x = xRow[ka];
    a.y = xRow[ka + 1];
    b.x = wrRow[ka];
    b.y = wrRow[ka + 1];
    acc = __builtin_amdgcn_wmma_f32_16x16x4_f32(false, a, false, b,
                                                (short)0, acc, false, false);
  }

  const float bb = bias[colB];
#pragma unroll
  for (int i = 0; i < 8; ++i) {
    const long long r = tm * 16 + (long long)(hi * 8 + i);
    float v = acc[i] + bb;
    if (RELU) v = fmaxf(v, 0.0f);
    out[r * COUT + colB] = v;
  }
}

// ---------------------------------------------------------------------------
// Host-side orchestration (graph-capture safe: only kernel launches on stream)
// ---------------------------------------------------------------------------
extern "C" void kernel_launch(void* const* d_in, const int* in_sizes, int n_in,
                              void* d_out, int out_size, void* d_ws, size_t ws_size,
                              hipStream_t stream) {
  const float*     x   = (const float*)d_in[0];
  const long long* ei  = (const long long*)d_in[1];   // int64 [2, E]
  // d_in[2] = edge_attr, unused by the reference
  const float*     W1l = (const float*)d_in[3];
  const float*     W1r = (const float*)d_in[4];
  const float*     b1  = (const float*)d_in[5];
  const float*     W2l = (const float*)d_in[6];
  const float*     W2r = (const float*)d_in[7];
  const float*     b2  = (const float*)d_in[8];
  float*           out = (float*)d_out;

  const long long nN = (long long)in_sizes[0] / 128;  // IN_C = 128
  const long long nE = (long long)in_sizes[1] / 2;
  const long long* src = ei;
  const long long* dst = ei + nE;

  // Workspace layout (floats): [deg: nN][h: nN*256][buf: nN*256]
  float* deg = (float*)d_ws;
  float* h   = deg + nN;
  float* buf = h + nN * 256;

  const int ZB = 256;

  // --- Layer 1 ---
  {
    long long n4 = nN / 4;
    zero_f4_kernel<<<(n4 + ZB - 1) / ZB, ZB, 0, stream>>>((float4*)deg, n4);
    long long m4 = nN * 128 / 4;
    zero_f4_kernel<<<(m4 + ZB - 1) / ZB, ZB, 0, stream>>>((float4*)buf, m4);
  }
  degree_kernel<<<(unsigned)((nE + ZB - 1) / ZB), ZB, 0, stream>>>(dst, deg, nE);
  {
    long long t = nE * 32;                            // 128 ch / 4
    scatter_kernel<32><<<(unsigned)((t + ZB - 1) / ZB), ZB, 0, stream>>>(
        x, src, dst, buf, nE);
  }
  {
    long long tiles = (nN / 16) * (256 / 16);
    unsigned blocks = (unsigned)((tiles + 7) / 8);
    sage_gemm_kernel<128, 256, true><<<blocks, 256, 0, stream>>>(
        buf, deg, x, W1l, W1r, b1, h, nN);
  }

  // --- Layer 2 ---
  {
    long long m4 = nN * 256 / 4;
    zero_f4_kernel<<<(unsigned)((m4 + ZB - 1) / ZB), ZB, 0, stream>>>((float4*)buf, m4);
  }
  {
    long long t = nE * 64;                            // 256 ch / 4
    scatter_kernel<64><<<(unsigned)((t + ZB - 1) / ZB), ZB, 0, stream>>>(
        h, src, dst, buf, nE);
  }
  {
    long long tiles = (nN / 16) * (128 / 16);
    unsigned blocks = (unsigned)((tiles + 7) / 8);
    sage_gemm_kernel<256, 128, false><<<blocks, 256, 0, stream>>>(
        buf, deg, h, W2l, W2r, b2, out, nN);
  }
}